// ParamModel_6734508720752
// MI455X (gfx1250) — compile-verified
//
#include <hip/hip_runtime.h>
#include <hip/hip_bf16.h>

// ---------------------------------------------------------------------------
// 10-layer LSTM (H=4) over T=262144 steps, software-pipelined across layers in
// a single wave32.  Gate GEMVs run on the CDNA5 matrix pipe via
// V_WMMA_F32_16X16X4_F32:  D(16x16) = A(16x4) x B(4x16) + C
//   A = input vector replicated across all 16 rows (documented A layout),
//   B = W^T preloaded once into registers,
//   C = bias splat (persistent) or the previous WMMA's D (Wih/Whh chaining).
// Every lane holds gates[lane & 15] -> lane-per-gate activations using the
// gfx1250-native v_tanh_f32 (fallback: v_exp_f32 + v_rcp_f32), cross-lane via
// ds_bpermute.  Scan split: guarded prologue / unguarded steady state /
// guarded epilogue, so the 262k-iteration hot loop carries zero edge selects.
// ---------------------------------------------------------------------------

typedef float v2f __attribute__((ext_vector_type(2)));
typedef float v8f __attribute__((ext_vector_type(8)));

namespace {
constexpr int T = 262144;
constexpr int L = 10;
constexpr float kLog2e = 1.4426950408889634f;

template <bool B> struct BoolC { static constexpr bool value = B; };
}

__device__ __forceinline__ float bperm(int byteIdx, float v) {
  return __int_as_float(__builtin_amdgcn_ds_bpermute(byteIdx, __float_as_int(v)));
}

__device__ __forceinline__ float fast_rcp(float v) {
  return __builtin_amdgcn_rcpf(v);
}

__device__ __forceinline__ float fast_exp2(float v) {
#if __has_builtin(__builtin_amdgcn_exp2f)
  return __builtin_amdgcn_exp2f(v);
#else
  return __expf(v * 0.6931471805599453f);
#endif
}

// gfx1250 has a native TANH transcendental (ISA 8.4: TRANS32 ops include TANH)
__device__ __forceinline__ float fast_tanh(float v) {
#if __has_builtin(__builtin_amdgcn_tanhf)
  return __builtin_amdgcn_tanhf(v);          // v_tanh_f32
#else
  // tanh(v) = 2/(1+exp(-2v)) - 1
  return __builtin_fmaf(
      fast_rcp(1.0f + fast_exp2(v * (-2.0f * kLog2e))), 2.0f, -1.0f);
#endif
}

__global__ __launch_bounds__(32) void lstm_wave_pipeline(
    const float* __restrict__ x,      // [T]
    const float* __restrict__ W_ih0,  // [16,1]
    const float* __restrict__ W_ih,   // [L-1,16,4]
    const float* __restrict__ W_hh,   // [L,16,4]
    const float* __restrict__ b_ih,   // [L,16]
    const float* __restrict__ b_hh,   // [L,16]
    const float* __restrict__ W_fc,   // [4,40]
    const float* __restrict__ b_fc,   // [4]
    float* __restrict__ out)          // [4]
{
  const int  lane = threadIdx.x;
  const int  n    = lane & 15;        // gate index this lane owns (N dim)
  const bool lo   = lane < 16;        // lane half -> which K pair of A/B
  const int  kA   = lo ? 0 : 2;       // K index of .x component
  const int  m    = lane & 3;         // H index this lane owns for c/h math
  const int  idx0 = lo ? 0 : 2;       // source lanes for rebuilding A from h

  // ds_bpermute byte indices (compiler folds +16/+32/+48 into inst offsets)
  const int biI  = m << 2;
  const int biF  = (m + 4) << 2;
  const int biG  = (m + 8) << 2;
  const int biO  = (m + 12) << 2;
  const int biA0 = idx0 << 2;
  const int biA1 = (idx0 + 1) << 2;

  // Per-lane activation constants.  Gates [8,12) are tanh; others sigmoid,
  // expressed through tanh:  sigmoid(x) = 0.5*tanh(0.5*x) + 0.5.
  const bool  isTanh = (n >= 8) && (n < 12);
  const float preScl = isTanh ? 1.0f : 0.5f;
  const float aMul   = isTanh ? 1.0f : 0.5f;
  const float aAdd   = isTanh ? 0.0f : 0.5f;

  // ---- Preload weights as WMMA B operands (B[k][n] = W[n][k]) -------------
  //   VGPR0: lanes 0-15 -> K=0, lanes 16-31 -> K=2 ; VGPR1: K=1 / K=3.
  v2f  whhB[L], wihB[L];
  v8f  biasC[L];                       // persistent bias splats (C operands)
  float hA0[L], hA1[L], hlane[L], cst[L];

#pragma unroll
  for (int l = 0; l < L; ++l) {
    whhB[l].x = W_hh[l * 64 + n * 4 + kA];
    whhB[l].y = W_hh[l * 64 + n * 4 + kA + 1];
    if (l == 0) {
      wihB[l].x = lo ? W_ih0[n] : 0.0f;  // scalar input: only K=0 column real
      wihB[l].y = 0.0f;
    } else {
      wihB[l].x = W_ih[(l - 1) * 64 + n * 4 + kA];
      wihB[l].y = W_ih[(l - 1) * 64 + n * 4 + kA + 1];
    }
    const float b = b_ih[l * 16 + n] + b_hh[l * 16 + n];
    biasC[l] = (v8f){b, b, b, b, b, b, b, b};
    hA0[l] = 0.0f; hA1[l] = 0.0f; hlane[l] = 0.0f; cst[l] = 0.0f;
  }

  // One pipelined step: layer l processes time t = s - l.  Descending layer
  // order => layer l reads layer l-1's h from the PREVIOUS iteration, so all
  // 10 layer chains are independent (10-way ILP hides VALU/DS/WMMA latency).
  auto do_step = [&](int s, float xv, auto GC) {
    constexpr bool GUARDED = decltype(GC)::value;
#pragma unroll
    for (int li = 0; li < L; ++li) {
      const int l = L - 1 - li;

      v2f a_h;  a_h.x = hA0[l];  a_h.y = hA1[l];
      v8f d1 = __builtin_amdgcn_wmma_f32_16x16x4_f32(
          false, a_h, false, whhB[l], (short)0, biasC[l], false, false);

      v2f a_in;
      if (l == 0) {
        a_in.x = lo ? xv : 0.0f;   // inp = (x,0,0,0); lanes>=16 see K=2 -> 0
        a_in.y = 0.0f;
      } else {
        a_in.x = hA0[l - 1];
        a_in.y = hA1[l - 1];
      }
      v8f d2 = __builtin_amdgcn_wmma_f32_16x16x4_f32(
          false, a_in, false, wihB[l], (short)0, d1, false, false);

      // Every lane holds gates[lane & 15] in all 8 D VGPRs.
      const float gate = d2[0];
      const float act  = __builtin_fmaf(fast_tanh(gate * preScl), aMul, aAdd);

      // gather i/f/g/o for H-index m (lanes 0-15 hold all 16 activations)
      const float iv = bperm(biI, act);
      const float fv = bperm(biF, act);
      const float gv = bperm(biG, act);
      const float ov = bperm(biO, act);

      const float cn = __builtin_fmaf(fv, cst[l], iv * gv);
      const float hn = ov * fast_tanh(cn);

      // rebuild A-layout broadcast of h_new for next iteration
      const float nA0 = bperm(biA0, hn);
      const float nA1 = bperm(biA1, hn);

      if (GUARDED) {
        const bool upd = ((unsigned)(s - l)) < (unsigned)T;
        cst[l]   = upd ? cn  : cst[l];
        hlane[l] = upd ? hn  : hlane[l];
        hA0[l]   = upd ? nA0 : hA0[l];
        hA1[l]   = upd ? nA1 : hA1[l];
      } else {
        cst[l] = cn; hlane[l] = hn; hA0[l] = nA0; hA1[l] = nA1;
      }
    }
  };

  float xv = x[0];
  int s = 0;

  // Pipeline fill: layers with s < l must hold their zero state.
#pragma clang loop unroll(disable)
  for (; s < L - 1; ++s) {
    const float xn = x[s + 1];             // s+1 <= L-1 << T, always valid
    do_step(s, xv, BoolC<true>{});
    xv = xn;
  }

  // Steady state: s in [L-1, T-1) -- no guards, unconditional next-x load.
  for (; s < T - 1; ++s) {
    const int pf = (s + 128 < T) ? (s + 128) : (T - 1);
    __builtin_prefetch(x + pf, 0, 0);      // global_prefetch_b8, stays ahead
    const float xn = x[s + 1];
    do_step(s, xv, BoolC<false>{});
    xv = xn;
  }

  // s == T-1: still fully valid for every layer; next x is past the end.
  do_step(T - 1, xv, BoolC<false>{});

  // Pipeline drain: layers with s - l >= T freeze their final state.
#pragma clang loop unroll(disable)
  for (s = T; s < T + L - 1; ++s) {
    do_step(s, 0.0f, BoolC<true>{});
  }

  // ---- Final FC: out = W_fc @ concat_l(h_l) + b_fc  (4 x 40) --------------
  float acc = b_fc[m];
#pragma unroll
  for (int l = 0; l < L; ++l) {
#pragma unroll
    for (int k = 0; k < 4; ++k) {
      acc = __builtin_fmaf(W_fc[m * 40 + l * 4 + k], bperm(k << 2, hlane[l]), acc);
    }
  }
  if (lane < 4) out[lane] = acc;
}

extern "C" void kernel_launch(void* const* d_in, const int* in_sizes, int n_in,
                              void* d_out, int out_size, void* d_ws, size_t ws_size,
                              hipStream_t stream) {
  (void)in_sizes; (void)n_in; (void)d_ws; (void)ws_size; (void)out_size;
  const float* x     = (const float*)d_in[0];
  const float* W_ih0 = (const float*)d_in[1];
  const float* W_ih  = (const float*)d_in[2];
  const float* W_hh  = (const float*)d_in[3];
  const float* b_ih  = (const float*)d_in[4];
  const float* b_hh  = (const float*)d_in[5];
  const float* W_fc  = (const float*)d_in[6];
  const float* b_fc  = (const float*)d_in[7];
  float* out = (float*)d_out;

  lstm_wave_pipeline<<<dim3(1), dim3(32), 0, stream>>>(
      x, W_ih0, W_ih, W_hh, b_ih, b_hh, W_fc, b_fc, out);
}